// SDFRenderer_13769665151086
// MI455X (gfx1250) — compile-verified
//
#include <hip/hip_runtime.h>

#define NS 96      // samples per ray
#define WPB 8      // waves (rays) per block

// ---- CDNA5 async LDS<->global ops (gfx1250; tracked by ASYNCcnt) ----------
__device__ __forceinline__ void async_ld_b128(unsigned lds_addr, const float* gaddr) {
    // copies 16 bytes from the global address in gaddr to the LDS address in lds_addr, per active lane
    asm volatile("global_load_async_to_lds_b128 %0, %1, off"
                 :: "v"(lds_addr), "v"(gaddr) : "memory");
}
__device__ __forceinline__ void async_st_b128(float* gaddr, unsigned lds_addr) {
    // copies 16 bytes from the LDS address in lds_addr to the global address in gaddr, per active lane
    asm volatile("global_store_async_from_lds_b128 %0, %1, off"
                 :: "v"(gaddr), "v"(lds_addr) : "memory");
}
__device__ __forceinline__ void wait_async0() {
    asm volatile("s_wait_asynccnt 0" ::: "memory");
}
__device__ __forceinline__ unsigned lds_off(const void* p) {
    // flat LDS address keeps the byte offset in addr[31:0]
    return (unsigned)(size_t)p;
}

__global__ void __launch_bounds__(WPB * 32)
sdf_render_kernel(const float* __restrict__ rgb,
                  const float* __restrict__ sdf,
                  const float* __restrict__ zv,
                  const int*   __restrict__ beta_p,
                  float* __restrict__ out, int N) {
    __shared__ alignas(16) float lsdf[WPB][NS];
    __shared__ alignas(16) float lz[WPB][NS];

    const int lane = threadIdx.x & 31;
    const int w    = threadIdx.x >> 5;
    const int ray  = blockIdx.x * WPB + w;
    if (ray >= N) return;                      // wave-uniform

    const size_t rbase = (size_t)ray * NS;
    const float* srow  = sdf + rbase;
    const float* zrow  = zv  + rbase;
    const float* crow  = rgb + rbase * 3;

    // ---- stage sdf & z rows into LDS via async DMA (384 B = 24 lanes x 16B)
    if (lane < 24) {
        async_ld_b128(lds_off(&lsdf[w][lane * 4]), srow + lane * 4);
        async_ld_b128(lds_off(&lz  [w][lane * 4]), zrow + lane * 4);
    }

    const float beta = (float)(*beta_p);       // scalar (s_load)

    // output layout: [depth N | rgb 3N | sdf 96N | z 96N]
    float* out_depth = out;
    float* out_rgb   = out + (size_t)N;
    float* out_sdf   = out + (size_t)4 * N;
    float* out_z     = out + (size_t)100 * N;

    wait_async0();                             // LDS rows now valid

    // ---- passthrough copy: LDS -> global, never touches VGPR data path
    if (lane < 24) {
        async_st_b128(out_sdf + rbase + lane * 4, lds_off(&lsdf[w][lane * 4]));
        async_st_b128(out_z   + rbase + lane * 4, lds_off(&lz  [w][lane * 4]));
    }

    // ---- per-ray exclusive cumprod scan + weighted reductions -------------
    float carry = 1.0f;                        // product of t over prior chunks
    float dep = 0.0f, r0 = 0.0f, r1 = 0.0f, r2 = 0.0f;

#pragma unroll
    for (int k = 0; k < 3; ++k) {
        const int s  = k * 32 + lane;          // sample index (consecutive across lanes)
        const float sd = lsdf[w][s];
        const float z  = lz[w][s];

        // alpha = 1 - exp(-beta * sigmoid(-sd*beta)); sigmoid(-x)=1/(1+e^x)
        const float e1    = __expf(sd * beta);
        const float sig   = __builtin_amdgcn_rcpf(1.0f + e1);
        const float e2    = __expf(-beta * sig);
        const float alpha = 1.0f - e2;
        const float t     = e2 + 1e-10f;       // 1 - alpha + 1e-10

        // inclusive product scan of t across the wave (Hillis-Steele)
        float p = t;
#pragma unroll
        for (int d = 1; d < 32; d <<= 1) {
            const float v = __shfl_up(p, d, 32);
            if (lane >= d) p *= v;
        }
        float excl = __shfl_up(p, 1, 32);      // exclusive within chunk
        if (lane == 0) excl = 1.0f;

        const float trans = carry * excl;      // global exclusive cumprod
        const float wgt   = alpha * trans;

        dep += wgt * z;
        const float c0 = crow[s * 3 + 0];
        const float c1 = crow[s * 3 + 1];
        const float c2 = crow[s * 3 + 2];
        r0 += wgt * c0;
        r1 += wgt * c1;
        r2 += wgt * c2;

        carry *= __shfl(p, 31, 32);            // fold chunk product into carry
    }

    // ---- wave reduction (sum over 96 samples) -----------------------------
#pragma unroll
    for (int off = 16; off > 0; off >>= 1) {
        dep += __shfl_down(dep, off, 32);
        r0  += __shfl_down(r0,  off, 32);
        r1  += __shfl_down(r1,  off, 32);
        r2  += __shfl_down(r2,  off, 32);
    }
    if (lane == 0) {
        out_depth[ray] = dep;
        out_rgb[(size_t)3 * ray + 0] = r0;
        out_rgb[(size_t)3 * ray + 1] = r1;
        out_rgb[(size_t)3 * ray + 2] = r2;
    }
    // async passthrough stores drain before S_ENDPGM (implicit wait-idle)
}

extern "C" void kernel_launch(void* const* d_in, const int* in_sizes, int n_in,
                              void* d_out, int out_size, void* d_ws, size_t ws_size,
                              hipStream_t stream) {
    const float* rgb  = (const float*)d_in[0];
    const float* sdf  = (const float*)d_in[1];
    const float* z    = (const float*)d_in[2];
    const int*   beta = (const int*)d_in[3];
    float* out = (float*)d_out;

    const int N = in_sizes[1] / NS;            // 65536 rays
    const int blocks = (N + WPB - 1) / WPB;    // 8192 blocks of 256 (8 waves)
    sdf_render_kernel<<<blocks, WPB * 32, 0, stream>>>(rgb, sdf, z, beta, out, N);
}